// Pooler_74543452389404
// MI455X (gfx1250) — compile-verified
//
#include <hip/hip_runtime.h>
#include <math.h>

#ifndef __has_builtin
#define __has_builtin(x) 0
#endif

#if defined(__HIP_DEVICE_COMPILE__) && \
    __has_builtin(__builtin_amdgcn_tensor_load_to_lds) && \
    __has_builtin(__builtin_amdgcn_s_wait_tensorcnt)
#define USE_TDM 1
#else
#define USE_TDM 0
#endif

#define THREADS       256
#define CELLS         125     // 5*5*5
#define NPTS          1000    // 10*10*10 sample points per ROI
#define CH_PER_STEP   2
#define CH_PER_BLOCK  16
#define NSTEPS        (CH_PER_BLOCK / CH_PER_STEP)   // 8
#define TILE_VOX      1536    // max voxels per channel staged in LDS

typedef unsigned int v4u __attribute__((ext_vector_type(4)));
typedef int          v8i __attribute__((ext_vector_type(8)));
typedef int          v4i __attribute__((ext_vector_type(4)));

#if USE_TDM
// Issue one TDM load: a 4D tile (t2 x t1 x t0 x CH_PER_STEP channels) of one
// feature volume (D x D x D per channel, channel stride D^3) into LDS.
// Descriptor packing per CDNA5 ISA sec 8.3-8.6 (D# groups 0..3).
// This toolchain's builtin takes 6 args: groups 0..3, an extended/reserved
// int32x8 group (zero-filled), and the cache-policy immediate.
__device__ __forceinline__ void tdm_issue_tile(const float* gbase, unsigned lds_off,
                                               int D, int C,
                                               int o0, int o1, int o2,
                                               int t0, int t1, int t2) {
  unsigned long long ga = (unsigned long long)(const void*)
      (gbase + (((long)o0 * D + o1) * (long)D + o2));
  unsigned long long s0 = (unsigned long long)D;                 // dim0 stride (elems)
  unsigned long long s1 = (unsigned long long)D * (unsigned)D;   // dim1 stride
  unsigned long long s2 = s1 * (unsigned)D;                      // dim2 stride (= channel)
  v4u g0; v8i g1; v4i g2; v4i g3; v8i g4;
  // ---- group 0: count=1, lds_addr, global_addr[56:0], type=2 ----
  g0[0] = 1u;
  g0[1] = lds_off;
  g0[2] = (unsigned)(ga & 0xFFFFFFFFull);
  g0[3] = (unsigned)((ga >> 32) & 0x01FFFFFFull) | 0x80000000u;  // type=2 at [127:126]
  // ---- group 1: data_size=4B(code 2); tensor_dim0/1=D; tile_dim0..2; strides ----
  g1[0] = (int)(2u << 16);                                             // data_size=2 (4B)
  g1[1] = (int)(((unsigned)D & 0xFFFFu) << 16);                        // tensor_dim0[15:0] @bit48
  g1[2] = (int)((((unsigned)D >> 16) & 0xFFFFu)                        // tensor_dim0[31:16]
              | (((unsigned)D & 0xFFFFu) << 16));                      // tensor_dim1[15:0]
  g1[3] = (int)((((unsigned)D >> 16) & 0xFFFFu)                        // tensor_dim1[31:16]
              | ((unsigned)t2 << 16));                                 // tile_dim0 = contig extent
  g1[4] = (int)((unsigned)t1 | ((unsigned)t0 << 16));                  // tile_dim1, tile_dim2
  g1[5] = (int)(unsigned)(s0 & 0xFFFFFFFFull);                         // tensor_dim0_stride lo32
  g1[6] = (int)((unsigned)((s0 >> 32) & 0xFFFFull)
              | (unsigned)((s1 & 0xFFFFull) << 16));                   // dim1_stride[15:0] @bit208
  g1[7] = (int)(unsigned)((s1 >> 16) & 0xFFFFFFFFull);                 // dim1_stride[47:16]
  // ---- group 2: tensor_dim2=D, tensor_dim3=C, dim2_stride=D^3, tile_dim3=2 ----
  g2[0] = D;
  g2[1] = C;
  g2[2] = (int)(unsigned)(s2 & 0xFFFFFFFFull);
  g2[3] = (int)((unsigned)((s2 >> 32) & 0xFFFFull)
              | ((unsigned)CH_PER_STEP << 16));
  // ---- group 3: unused (5th dim) ----
  g3[0] = 0; g3[1] = 0; g3[2] = 0; g3[3] = 0;
  // ---- extended/reserved group (6-arg builtin form): zero ----
  g4[0] = 0; g4[1] = 0; g4[2] = 0; g4[3] = 0;
  g4[4] = 0; g4[5] = 0; g4[6] = 0; g4[7] = 0;
  __builtin_amdgcn_tensor_load_to_lds(g0, g1, g2, g3, g4, 0);
}
#endif

__global__ __launch_bounds__(THREADS)
void Pooler_74543452389404_kernel(const float* __restrict__ x0,
                                  const float* __restrict__ x1,
                                  const float* __restrict__ x2,
                                  const float* __restrict__ x3,
                                  const float* __restrict__ bbox,
                                  const int*   __restrict__ bids,
                                  float*       __restrict__ out,
                                  int C) {
  __shared__ unsigned s_pk[NPTS];                 // i0|i1<<8|i2<<16|steps<<24|valid<<27
  __shared__ float    s_f0[NPTS], s_f1[NPTS], s_f2[NPTS];
  __shared__ int      s_min[3], s_max[3];
  __shared__ int      s_prm[8];                   // o0,o1,o2,t0,t1,t2,tvox,use
  __shared__ float    s_tile[2][CH_PER_STEP * TILE_VOX];

  const int tid = threadIdx.x;
  const int roi = blockIdx.x;
  const int cg  = blockIdx.y;                     // channel group (16 channels)

  // ---- per-ROI uniform setup (redundant per thread; 8 scalar-ish loads) ----
  const float* bb = bbox + (size_t)roi * 7;
  const float b0 = bb[0], b1 = bb[1], b2 = bb[2];
  const float b3 = bb[3], b4 = bb[4], b5 = bb[5], b6 = bb[6];
  const int   bid = bids[roi];

  const float SC[4] = {0.25f, 0.125f, 0.0625f, 0.03125f};
  const float rate = sqrtf(fmaxf(b3, b4)) * 0.05f;      // sqrt(max(dx,dy))/20
  int lvl = 0; float best = fabsf(SC[0] - rate);
  #pragma unroll
  for (int l = 1; l < 4; ++l) {
    float d = fabsf(SC[l] - rate);
    if (d < best) { best = d; lvl = l; }                // argmin, first-min wins
  }
  const float scale = SC[lvl];
  const int   D     = 64 >> lvl;
  const float Df    = (float)D;
  const float* feat = x0;
  if      (lvl == 1) feat = x1;
  else if (lvl == 2) feat = x2;
  else if (lvl == 3) feat = x3;
  const size_t vol = (size_t)D * D * D;
  const float* fbase = feat + ((size_t)bid * C + (size_t)cg * CH_PER_BLOCK) * vol;

  // roi7 = [b1,b0,b2,b4,b3,b5]; ctr = roi7[0:3]*scale; size = roi7[3:6]*scale
  const float ctr0 = b1 * scale, ctr1 = b0 * scale, ctr2 = b2 * scale;
  const float sz0  = b4 * scale, sz1  = b3 * scale, sz2  = b5 * scale;
  const float ct = cosf(b6), st = sinf(b6);

  if (tid < 3) { s_min[tid] = 1 << 30; s_max[tid] = -1; }
  __syncthreads();

  // ---- phase 1: per-sample-point geometry table + index bounding box ----
  for (int p = tid; p < NPTS; p += THREADS) {
    const int a = p / 100, bq = (p / 10) % 10, kq = p % 10;
    // axis_samples(5,2): (bin + (sub+0.5)/2)/5, idx = bin*2+sub
    const float A0 = ((float)(a  >> 1) + ((a  & 1) + 0.5f) * 0.5f) * 0.2f;
    const float A1 = ((float)(bq >> 1) + ((bq & 1) + 0.5f) * 0.5f) * 0.2f;
    const float A2 = ((float)(kq >> 1) + ((kq & 1) + 0.5f) * 0.5f) * 0.2f;
    const float O0 = (A0 - 0.5f) * sz0;
    const float O1 = (A1 - 0.5f) * sz1;
    const float O2 = (A2 - 0.5f) * sz2;
    const float q0 = ctr0 + ct * O0 - st * O1;
    const float q1 = ctr1 + st * O0 + ct * O1;
    const float q2 = ctr2 + O2;
    const bool valid = (q0 > -1.f) && (q0 < Df) && (q1 > -1.f) && (q1 < Df) &&
                       (q2 > -1.f) && (q2 < Df);
    const float p0 = fminf(fmaxf(q0, 0.f), Df - 1.f);
    const float p1 = fminf(fmaxf(q1, 0.f), Df - 1.f);
    const float p2 = fminf(fmaxf(q2, 0.f), Df - 1.f);
    const float fl0 = floorf(p0), fl1 = floorf(p1), fl2 = floorf(p2);
    const int i0 = (int)fl0, i1 = (int)fl1, i2 = (int)fl2;
    const int st0 = (i0 < D - 1), st1 = (i1 < D - 1), st2 = (i2 < D - 1);
    s_pk[p] = (unsigned)i0 | ((unsigned)i1 << 8) | ((unsigned)i2 << 16) |
              ((unsigned)st0 << 24) | ((unsigned)st1 << 25) |
              ((unsigned)st2 << 26) | ((unsigned)(valid ? 1 : 0) << 27);
    s_f0[p] = p0 - fl0; s_f1[p] = p1 - fl1; s_f2[p] = p2 - fl2;
    atomicMin(&s_min[0], i0); atomicMax(&s_max[0], i0 + st0);
    atomicMin(&s_min[1], i1); atomicMax(&s_max[1], i1 + st1);
    atomicMin(&s_min[2], i2); atomicMax(&s_max[2], i2 + st2);
  }
  __syncthreads();

  if (tid == 0) {
    const int o0 = s_min[0], o1 = s_min[1], o2 = s_min[2];
    const int t0 = s_max[0] - o0 + 1;
    const int t1 = s_max[1] - o1 + 1;
    const int t2 = s_max[2] - o2 + 1;
    const int tvox = t0 * t1 * t2;
    int use = (tvox > 0 && tvox <= TILE_VOX) ? 1 : 0;
#if !USE_TDM
    use = 0;
#endif
    s_prm[0] = o0; s_prm[1] = o1; s_prm[2] = o2;
    s_prm[3] = t0; s_prm[4] = t1; s_prm[5] = t2;
    s_prm[6] = tvox; s_prm[7] = use;
  }
  __syncthreads();

  const int o0 = s_prm[0], o1 = s_prm[1], o2 = s_prm[2];
  const int t0v = s_prm[3], t1v = s_prm[4], t2v = s_prm[5];
  const int tvox = s_prm[6];
  const int use  = s_prm[7];

  // ---- phase 2: per-(cell, channel) trilinear pooling ----
  auto process = [&](int s, int bufIdx, bool tilePath) {
    const int t = tid;
    if (t < CELLS * CH_PER_STEP) {
      const int chl  = t / CELLS;
      const int cell = t - chl * CELLS;
      const int ci = cell / 25, cj = (cell / 5) % 5, ck = cell % 5;
      const float* gb = fbase + (size_t)(s * CH_PER_STEP + chl) * vol;
      const float* tb = &s_tile[bufIdx][chl * tvox];
      if (!tilePath) __builtin_prefetch(gb, 0, 1);     // global_prefetch_b8
      float acc = 0.f;
      #pragma unroll
      for (int u = 0; u < 2; ++u)
      #pragma unroll
      for (int v = 0; v < 2; ++v)
      #pragma unroll
      for (int w = 0; w < 2; ++w) {
        const int p = ((2 * ci + u) * 10 + (2 * cj + v)) * 10 + (2 * ck + w);
        const unsigned pk = s_pk[p];
        const float f0 = s_f0[p], f1 = s_f1[p], f2 = s_f2[p];
        const float validf = (float)((pk >> 27) & 1u);
        const int i0 = (int)(pk & 255u);
        const int i1 = (int)((pk >> 8) & 255u);
        const int i2 = (int)((pk >> 16) & 255u);
        float v000, v001, v010, v011, v100, v101, v110, v111;
        if (tilePath) {
          const int b  = ((i0 - o0) * t1v + (i1 - o1)) * t2v + (i2 - o2);
          const int sA = ((pk >> 24) & 1u) ? t1v * t2v : 0;
          const int sB = ((pk >> 25) & 1u) ? t2v : 0;
          const int sC = (int)((pk >> 26) & 1u);
          v000 = tb[b];          v001 = tb[b + sC];
          v010 = tb[b + sB];     v011 = tb[b + sB + sC];
          v100 = tb[b + sA];     v101 = tb[b + sA + sC];
          v110 = tb[b + sA + sB]; v111 = tb[b + sA + sB + sC];
        } else {
          const size_t b = ((size_t)i0 * D + i1) * D + i2;
          const int sA = ((pk >> 24) & 1u) ? D * D : 0;
          const int sB = ((pk >> 25) & 1u) ? D : 0;
          const int sC = (int)((pk >> 26) & 1u);
          v000 = gb[b];          v001 = gb[b + sC];
          v010 = gb[b + sB];     v011 = gb[b + sB + sC];
          v100 = gb[b + sA];     v101 = gb[b + sA + sC];
          v110 = gb[b + sA + sB]; v111 = gb[b + sA + sB + sC];
        }
        const float w0 = 1.f - f0, w1 = 1.f - f1, w2 = 1.f - f2;
        const float c00 = v000 * w2 + v001 * f2;
        const float c01 = v010 * w2 + v011 * f2;
        const float c10 = v100 * w2 + v101 * f2;
        const float c11 = v110 * w2 + v111 * f2;
        const float r  = (c00 * w1 + c01 * f1) * w0 + (c10 * w1 + c11 * f1) * f0;
        acc += r * validf;
      }
      const int gc = cg * CH_PER_BLOCK + s * CH_PER_STEP + chl;
      out[((size_t)roi * C + gc) * CELLS + cell] = acc * 0.125f;
    }
  };

#if USE_TDM
  if (use) {
    unsigned ldsbase[2];
    ldsbase[0] = (unsigned)(unsigned long long)(const void*)&s_tile[0][0];
    ldsbase[1] = (unsigned)(unsigned long long)(const void*)&s_tile[1][0];
    if (tid < 32) {   // wave 0 drives the TDM pipeline
      tdm_issue_tile(fbase, ldsbase[0], D, C, o0, o1, o2, t0v, t1v, t2v);
    }
    for (int s = 0; s < NSTEPS; ++s) {
      if (tid < 32) {
        if (s + 1 < NSTEPS) {
          tdm_issue_tile(fbase + (size_t)((s + 1) * CH_PER_STEP) * vol,
                         ldsbase[(s + 1) & 1], D, C, o0, o1, o2, t0v, t1v, t2v);
          __builtin_amdgcn_s_wait_tensorcnt(1);   // tile s complete
        } else {
          __builtin_amdgcn_s_wait_tensorcnt(0);
        }
      }
      __syncthreads();          // publish tile s to all waves
      process(s, s & 1, true);
      __syncthreads();          // done reading buf before tile s+2 overwrites it
    }
  } else
#endif
  {
    for (int s = 0; s < NSTEPS; ++s) process(s, 0, false);
  }
}

extern "C" void kernel_launch(void* const* d_in, const int* in_sizes, int n_in,
                              void* d_out, int out_size, void* d_ws, size_t ws_size,
                              hipStream_t stream) {
  (void)n_in; (void)out_size; (void)d_ws; (void)ws_size;
  const float* x0   = (const float*)d_in[0];
  const float* x1   = (const float*)d_in[1];
  const float* x2   = (const float*)d_in[2];
  const float* x3   = (const float*)d_in[3];
  const float* bbox = (const float*)d_in[4];
  const int*   bids = (const int*)d_in[5];
  float* out = (float*)d_out;

  const int R = in_sizes[4] / 7;        // 128 ROIs
  const int C = 64;                     // channels (B=2, 64x64^3 per reference)
  dim3 grid(R, C / CH_PER_BLOCK);       // (128, 4)
  Pooler_74543452389404_kernel<<<grid, THREADS, 0, stream>>>(
      x0, x1, x2, x3, bbox, bids, out, C);
}